// SIGN_3135326126434
// MI455X (gfx1250) — compile-verified
//
#include <hip/hip_runtime.h>

#define N_NODES 50000
#define N_EDGES 800000
#define N_HOPS  4
#define NFEAT   256
#define NHID    128
#define NOUT    64
#define KCAT    (N_HOPS * NHID)   // 512

#define LDSW (NHID + 4)           // padded LDS stride for W  (132)
#define LDSO (NOUT + 4)           // padded LDS stride for W_out (68)

typedef __attribute__((ext_vector_type(2))) float v2f;
typedef __attribute__((ext_vector_type(8))) float v8f;

__device__ __forceinline__ v8f wmma_f32_16x16x4(v2f a, v2f b, v8f c) {
    // 8 args: (neg_a, A, neg_b, B, c_mod, C, reuse_a, reuse_b)
    return __builtin_amdgcn_wmma_f32_16x16x4_f32(
        false, a, false, b, (short)0, c, false, false);
}

// ---------------------------------------------------------------------------
// Kernel 1: h[k] = x @ W[k] + b[k]   for k = blockIdx.y (4 hops)
//   wave -> one 16-row slab of M, all 8 N-tiles (NHID=128)
//   W[k] (256x128 f32) staged in dynamic LDS with padded stride
// ---------------------------------------------------------------------------
__global__ __launch_bounds__(256) void sign_hop_gemm(
    const float* __restrict__ x, const float* __restrict__ W,
    const float* __restrict__ b, float* __restrict__ h)
{
    extern __shared__ float Wl[];                 // NFEAT * LDSW floats
    const int hop = blockIdx.y;
    const float* Wk = W + (size_t)hop * NFEAT * NHID;

    // cooperative staged load of W[k] into LDS (float4 granularity)
    for (int i = threadIdx.x * 4; i < NFEAT * NHID; i += 256 * 4) {
        float4 w4 = *(const float4*)(Wk + i);
        const int r = i >> 7;            // / NHID
        const int c = i & (NHID - 1);
        float* d = Wl + r * LDSW + c;
        d[0] = w4.x; d[1] = w4.y; d[2] = w4.z; d[3] = w4.w;
    }
    __syncthreads();

    const int wave = threadIdx.x >> 5;
    const int lane = threadIdx.x & 31;
    const int ml   = lane & 15;          // row within M-tile (A) / col (B,C,D)
    const int kh   = lane >> 4;          // K-half select
    const int m0   = (blockIdx.x * 8 + wave) * 16;
    if (m0 >= N_NODES) return;           // 50000 % 16 == 0 -> tile-level check

    v8f acc[8] = {};
    const float* xr = x + (size_t)(m0 + ml) * NFEAT + 2 * kh;

    for (int k0 = 0; k0 < NFEAT; k0 += 4) {
        v2f a; a.x = xr[k0]; a.y = xr[k0 + 1];
        const float* wl0 = Wl + (k0 + 2 * kh) * LDSW + ml;
#pragma unroll
        for (int t = 0; t < 8; t++) {
            v2f bb; bb.x = wl0[t * 16]; bb.y = wl0[LDSW + t * 16];
            acc[t] = wmma_f32_16x16x4(a, bb, acc[t]);
        }
    }

    float* ho = h + (size_t)hop * N_NODES * NHID;
    const float* bk = b + hop * NHID;
#pragma unroll
    for (int t = 0; t < 8; t++) {
        const int n = t * 16 + ml;
        const float bias = bk[n];
#pragma unroll
        for (int v = 0; v < 8; v++) {
            const int m = m0 + v + 8 * kh;
            ho[(size_t)m * NHID + n] = acc[t][v] + bias;
        }
    }
}

// ---------------------------------------------------------------------------
// Kernel 2: zero the aggregation buffer (float4 granularity)
// ---------------------------------------------------------------------------
__global__ __launch_bounds__(256) void sign_zero4(float4* __restrict__ p, long long n4)
{
    const long long i = (long long)blockIdx.x * blockDim.x + threadIdx.x;
    if (i < n4) p[i] = make_float4(0.f, 0.f, 0.f, 0.f);
}

// ---------------------------------------------------------------------------
// Kernel 3: COO SpMM scatter  agg[k][row] += val * h[k][col]
//   one wave per 4 edges; each lane owns a float4 chunk of the 128-wide row
// ---------------------------------------------------------------------------
__global__ __launch_bounds__(256) void sign_spmm(
    const int*   __restrict__ rows, const int* __restrict__ cols,
    const float* __restrict__ vals, const float* __restrict__ h,
    float* __restrict__ agg)
{
    const int lane = threadIdx.x & 31;
    const long long wid  = (long long)blockIdx.x * 8 + (threadIdx.x >> 5);
    const long long base = wid * 4;                       // first edge (flat)
    const long long total = (long long)N_HOPS * N_EDGES;
    if (base >= total) return;

    // N_EDGES % 4 == 0 -> a 4-edge group never crosses a hop boundary
    const int k  = (int)(base / N_EDGES);
    const int e0 = (int)(base - (long long)k * N_EDGES);
    const size_t eb = (size_t)k * N_EDGES;
    const float* hk = h   + (size_t)k * N_NODES * NHID;
    float*       ak = agg + (size_t)k * N_NODES * NHID;

#pragma unroll
    for (int i = 0; i < 4; i++) {
        const int   e = e0 + i;
        const int   r = rows[eb + e];
        const int   c = cols[eb + e];
        const float v = vals[eb + e];
        const float4 hv = *(const float4*)(hk + (size_t)c * NHID + lane * 4);
        float* dst = ak + (size_t)r * NHID + lane * 4;
        atomicAdd(dst + 0, v * hv.x);
        atomicAdd(dst + 1, v * hv.y);
        atomicAdd(dst + 2, v * hv.z);
        atomicAdd(dst + 3, v * hv.w);
    }
}

// ---------------------------------------------------------------------------
// Kernel 4: out = elu(concat(agg)) @ W_out + b_out
//   concat[n, c] = agg[c>>7][n][c&127]; ELU fused into the A-fragment load
//   W_out (512x64 f32) staged in dynamic LDS
// ---------------------------------------------------------------------------
__global__ __launch_bounds__(256) void sign_out_gemm(
    const float* __restrict__ agg, const float* __restrict__ Wout,
    const float* __restrict__ bout, float* __restrict__ out)
{
    extern __shared__ float Wl[];                 // KCAT * LDSO floats
    for (int i = threadIdx.x * 4; i < KCAT * NOUT; i += 256 * 4) {
        float4 w4 = *(const float4*)(Wout + i);
        const int r = i >> 6;            // / NOUT
        const int c = i & (NOUT - 1);
        float* d = Wl + r * LDSO + c;
        d[0] = w4.x; d[1] = w4.y; d[2] = w4.z; d[3] = w4.w;
    }
    __syncthreads();

    const int wave = threadIdx.x >> 5;
    const int lane = threadIdx.x & 31;
    const int ml   = lane & 15;
    const int kh   = lane >> 4;
    const int m0   = (blockIdx.x * 8 + wave) * 16;
    if (m0 >= N_NODES) return;

    v8f acc[4] = {};
    const int m = m0 + ml;

    for (int k0 = 0; k0 < KCAT; k0 += 4) {
        const int c    = k0 + 2 * kh;             // even -> c, c+1 same hop block
        const int hopI = c >> 7;
        const int off  = c & 127;
        const float* ap = agg + ((size_t)hopI * N_NODES + m) * NHID + off;
        float a0 = ap[0], a1 = ap[1];
        a0 = a0 > 0.f ? a0 : (__expf(a0) - 1.f);  // ELU, alpha = 1
        a1 = a1 > 0.f ? a1 : (__expf(a1) - 1.f);
        v2f a; a.x = a0; a.y = a1;
        const float* wl0 = Wl + c * LDSO + ml;
#pragma unroll
        for (int t = 0; t < 4; t++) {
            v2f bb; bb.x = wl0[t * 16]; bb.y = wl0[LDSO + t * 16];
            acc[t] = wmma_f32_16x16x4(a, bb, acc[t]);
        }
    }

#pragma unroll
    for (int t = 0; t < 4; t++) {
        const int n = t * 16 + ml;
        const float bias = bout[n];
#pragma unroll
        for (int v = 0; v < 8; v++) {
            const int mm = m0 + v + 8 * kh;
            out[(size_t)mm * NOUT + n] = acc[t][v] + bias;
        }
    }
}

// ---------------------------------------------------------------------------
extern "C" void kernel_launch(void* const* d_in, const int* in_sizes, int n_in,
                              void* d_out, int out_size, void* d_ws, size_t ws_size,
                              hipStream_t stream)
{
    (void)in_sizes; (void)n_in; (void)out_size; (void)ws_size;

    const float* x     = (const float*)d_in[0];
    const float* W     = (const float*)d_in[1];
    const float* b     = (const float*)d_in[2];
    const float* W_out = (const float*)d_in[3];
    const float* b_out = (const float*)d_in[4];
    const int*   erow  = (const int*)  d_in[5];
    const int*   ecol  = (const int*)  d_in[6];
    const float* eval  = (const float*)d_in[7];
    float* out = (float*)d_out;

    float* h   = (float*)d_ws;                                   // 4*50000*128 f32
    float* agg = h + (size_t)N_HOPS * N_NODES * NHID;            // 4*50000*128 f32

    // 1) per-hop linear (WMMA, W[k] in dynamic LDS)
    {
        dim3 g((N_NODES + 127) / 128, N_HOPS);
        sign_hop_gemm<<<g, 256, NFEAT * LDSW * sizeof(float), stream>>>(x, W, b, h);
    }

    // 2) zero agg (graph replays: must re-zero every call)
    {
        const long long n4 = (long long)N_HOPS * N_NODES * NHID / 4;
        const unsigned blocks = (unsigned)((n4 + 255) / 256);
        sign_zero4<<<blocks, 256, 0, stream>>>((float4*)agg, n4);
    }

    // 3) COO SpMM scatter with f32 atomics
    {
        const long long waves  = (long long)N_HOPS * N_EDGES / 4;   // 800000
        const unsigned  blocks = (unsigned)((waves + 7) / 8);
        sign_spmm<<<blocks, 256, 0, stream>>>(erow, ecol, eval, h, agg);
    }

    // 4) fused ELU + output GEMM (WMMA, W_out in dynamic LDS)
    {
        const unsigned blocks = (N_NODES + 127) / 128;
        sign_out_gemm<<<blocks, 256, KCAT * LDSO * sizeof(float), stream>>>(
            agg, W_out, b_out, out);
    }
}